// mil_outputs_44736379355483
// MI455X (gfx1250) — compile-verified
//
#include <hip/hip_runtime.h>
#include <hip/hip_bf16.h>

// GAT-based MIL scoring for MI455X (gfx1250, wave32, WMMA).
// Heavy GEMMs run on V_WMMA_F32_16X16X4_F32 with LDS-staged, pre-swizzled
// A/B operands. er = x @ (Wr^T att_r) folding removes half the GEMM work.

typedef __attribute__((ext_vector_type(2))) float v2f;
typedef __attribute__((ext_vector_type(8))) float v8f;

static constexpr int N_NODES = 50000;
static constexpr int N_EDGES = 500000;
static constexpr int DIN  = 512;
static constexpr int HID  = 128;
static constexpr int DOUT = 20;
static constexpr int MTILES = N_NODES / 16;  // 3125 (exact)
#define NEG_SLOPE 0.2f

// ---------------------------------------------------------------------------
// WMMA GEMM: Out[M x CVALID] = A[M x K] @ W[CVALID x K]^T   (f32, 16x16x4)
// Block = CT waves; wave t owns output col-tile t. Whole W staged in LDS in
// B-operand order; each 16-row A subtile staged in A-operand order.
// ---------------------------------------------------------------------------
template<int K, int CT, int CVALID>
__global__ void wmma_gemm(const float* __restrict__ A,
                          const float* __restrict__ W,
                          float* __restrict__ Out,
                          int m_tiles_total, int mt_per_block) {
  extern __shared__ float smem[];
  constexpr int KG = K / 4;                 // K-groups of 4
  float* sB = smem;                         // CT*K*16 floats (B operand order)
  float* sA = smem + CT * K * 16;           // K*16 floats   (A operand order)
  const int tid  = threadIdx.x;
  const int nthr = CT * 32;
  const int lane = tid & 31;
  const int wave = tid >> 5;

  // Zero-pad then stage W (coalesced over W's row-major storage).
  for (int i = tid; i < CT * K * 16; i += nthr) sB[i] = 0.0f;
  __syncthreads();
  for (int i = tid; i < CVALID * K; i += nthr) {
    int row = i / K, k = i - row * K;       // row = output col, k = reduction
    int t = row >> 4, rl = row & 15;
    int g = k >> 2, kin = k & 3;
    int l = ((kin >> 1) << 4) | rl;         // lanes 16-31 hold k%4 in {2,3}
    sB[((t * KG + g) * 32 + l) * 2 + (kin & 1)] = W[i];
  }
  __syncthreads();

  const float* bBase = sB + (size_t)wave * KG * 64;
  const int mt0 = blockIdx.x * mt_per_block;
  for (int mi = 0; mi < mt_per_block; ++mi) {
    int mt = mt0 + mi;
    if (mt >= m_tiles_total) break;         // block-uniform
    const float* Ab = A + (size_t)mt * 16 * K;
    for (int i = tid; i < 16 * K; i += nthr) {
      int r = i / K, k = i - r * K;
      int g = k >> 2, kin = k & 3;
      int l = ((kin >> 1) << 4) | r;
      sA[(g * 32 + l) * 2 + (kin & 1)] = Ab[i];
    }
    __syncthreads();

    v8f acc = {};
    #pragma unroll 4
    for (int g = 0; g < KG; ++g) {
      v2f a = *(const v2f*)(sA + g * 64 + lane * 2);
      v2f b = *(const v2f*)(bBase + g * 64 + lane * 2);
      acc = __builtin_amdgcn_wmma_f32_16x16x4_f32(
          false, a, false, b, (short)0, acc, false, false);
    }

    // D layout: VGPR r -> (M=r, N=lane) lanes 0-15; (M=r+8, N=lane-16) hi.
    int col   = wave * 16 + (lane & 15);
    int mhalf = (lane >> 4) * 8;
    if (col < CVALID) {
      float* o = Out + ((size_t)mt * 16 + mhalf) * CVALID + col;
      #pragma unroll
      for (int r = 0; r < 8; ++r) o[(size_t)r * CVALID] = acc[r];
    }
    __syncthreads();
  }
}

// ---------------------------------------------------------------------------
// vr[k] = sum_j Wr[j][k] * ar[j]   (folds the xr GEMM into a vector)
// ---------------------------------------------------------------------------
__global__ void make_vr(const float* __restrict__ Wr, const float* __restrict__ ar,
                        float* __restrict__ vr, int K, int J) {
  int k = blockIdx.x * blockDim.x + threadIdx.x;
  if (k >= K) return;
  float s = 0.0f;
  for (int j = 0; j < J; ++j) s += Wr[(size_t)j * K + k] * ar[j];
  vr[k] = s;
}

// el[n] = xl[n,:dimL] . al ; er[n] = srcR[n,:dimR] . vr   (one wave per node)
__global__ void attn_scalars(const float* __restrict__ srcR, int dimR,
                             const float* __restrict__ vr,
                             const float* __restrict__ xl, int dimL,
                             const float* __restrict__ al,
                             float* __restrict__ el, float* __restrict__ er, int n) {
  int node = blockIdx.x * (blockDim.x >> 5) + (threadIdx.x >> 5);
  int lane = threadIdx.x & 31;
  if (node >= n) return;
  const float* rr = srcR + (size_t)node * dimR;
  const float* lr = xl   + (size_t)node * dimL;
  float se = 0.0f, sl = 0.0f;
  for (int k = lane; k < dimR; k += 32) se += rr[k] * vr[k];
  for (int k = lane; k < dimL; k += 32) sl += lr[k] * al[k];
  for (int off = 16; off; off >>= 1) {
    se += __shfl_down(se, off);
    sl += __shfl_down(sl, off);
  }
  if (lane == 0) { er[node] = se; el[node] = sl; }
}

// -------------------------- edge-softmax phase -----------------------------
__device__ __forceinline__ float leaky(float x) {
  return x > 0.0f ? x : NEG_SLOPE * x;
}
__device__ __forceinline__ unsigned enc_f32(float f) {
  unsigned u = __float_as_uint(f);
  return (u & 0x80000000u) ? ~u : (u | 0x80000000u);
}

__global__ void edge_max(const long long* __restrict__ ei,
                         const float* __restrict__ el, const float* __restrict__ er,
                         unsigned* __restrict__ emaxEnc, int E) {
  int i = blockIdx.x * blockDim.x + threadIdx.x;
  if (i >= E) return;
  int s = (int)ei[i], d = (int)ei[(size_t)E + i];
  float e = leaky(el[s] + er[d]);
  atomicMax(&emaxEnc[d], enc_f32(e));
}

__global__ void decode_max(const unsigned* __restrict__ enc,
                           float* __restrict__ emax, int n) {
  int i = blockIdx.x * blockDim.x + threadIdx.x;
  if (i >= n) return;
  unsigned u = enc[i];
  float f = (u & 0x80000000u) ? __uint_as_float(u ^ 0x80000000u)
                              : __uint_as_float(~u);
  emax[i] = (f < -1e37f) ? 0.0f : f;  // isolated node -> 0 (matches reference)
}

__global__ void edge_expsum(const long long* __restrict__ ei,
                            const float* __restrict__ el, const float* __restrict__ er,
                            const float* __restrict__ emax,
                            float* __restrict__ wbuf, float* __restrict__ denom, int E) {
  int i = blockIdx.x * blockDim.x + threadIdx.x;
  if (i >= E) return;
  int s = (int)ei[i], d = (int)ei[(size_t)E + i];
  float w = expf(leaky(el[s] + er[d]) - emax[d]);
  wbuf[i] = w;
  atomicAdd(&denom[d], w);
}

// out[dst,:] += xl[src,:] * alpha  — CP threads per edge (C valid columns)
__global__ void edge_aggregate(const long long* __restrict__ ei,
                               const float* __restrict__ wbuf,
                               const float* __restrict__ denom,
                               const float* __restrict__ xl,
                               float* __restrict__ out, int E, int C, int CP) {
  int flat = blockIdx.x * blockDim.x + threadIdx.x;
  int i = flat / CP, c = flat - i * CP;
  if (i >= E || c >= C) return;
  int s = (int)ei[i], d = (int)ei[(size_t)E + i];
  float alpha = wbuf[i] / (denom[d] + 1e-16f);
  atomicAdd(&out[(size_t)d * C + c], xl[(size_t)s * C + c] * alpha);
}

// ------------------------------ elementwise --------------------------------
__global__ void fill_f32(float* p, float v, int n) {
  int i = blockIdx.x * blockDim.x + threadIdx.x;
  if (i < n) p[i] = v;
}
__global__ void fill_u32(unsigned* p, unsigned v, int n) {
  int i = blockIdx.x * blockDim.x + threadIdx.x;
  if (i < n) p[i] = v;
}
__global__ void init_bias(float* __restrict__ out, const float* __restrict__ bias,
                          int total, int C) {
  int i = blockIdx.x * blockDim.x + threadIdx.x;
  if (i < total) out[i] = bias[i % C];
}
__global__ void elu_inplace(float* p, int n) {
  int i = blockIdx.x * blockDim.x + threadIdx.x;
  if (i < n) { float v = p[i]; p[i] = v > 0.0f ? v : (expf(v) - 1.0f); }
}

// --------------------------- final dual softmax ----------------------------
__global__ void col_stats(const float* __restrict__ s0,
                          float* __restrict__ colmax, float* __restrict__ colsum,
                          int n, int C) {
  int c = blockIdx.x;                       // one block per class column
  __shared__ float red[256];
  float m = -3.4e38f;
  for (int i = threadIdx.x; i < n; i += blockDim.x)
    m = fmaxf(m, s0[(size_t)i * C + c]);
  red[threadIdx.x] = m; __syncthreads();
  for (int s = 128; s; s >>= 1) {
    if (threadIdx.x < s) red[threadIdx.x] = fmaxf(red[threadIdx.x], red[threadIdx.x + s]);
    __syncthreads();
  }
  float cm = red[0]; __syncthreads();
  float acc = 0.0f;
  for (int i = threadIdx.x; i < n; i += blockDim.x)
    acc += expf(s0[(size_t)i * C + c] - cm);
  red[threadIdx.x] = acc; __syncthreads();
  for (int s = 128; s; s >>= 1) {
    if (threadIdx.x < s) red[threadIdx.x] += red[threadIdx.x + s];
    __syncthreads();
  }
  if (threadIdx.x == 0) { colmax[c] = cm; colsum[c] = red[0]; }
}

__global__ void final_out(const float* __restrict__ s0, const float* __restrict__ s1,
                          const float* __restrict__ colmax,
                          const float* __restrict__ colsum,
                          float* __restrict__ out, int n, int C) {
  int i = blockIdx.x * blockDim.x + threadIdx.x;
  if (i >= n) return;
  const float* r1 = s1 + (size_t)i * C;
  const float* r0 = s0 + (size_t)i * C;
  float rmax = -3.4e38f;
  for (int c = 0; c < C; ++c) rmax = fmaxf(rmax, r1[c]);
  float rsum = 0.0f;
  for (int c = 0; c < C; ++c) rsum += expf(r1[c] - rmax);
  float* o = out + (size_t)i * C;
  for (int c = 0; c < C; ++c) {
    float sm0 = expf(r0[c] - colmax[c]) / colsum[c];
    float sm1 = expf(r1[c] - rmax) / rsum;
    o[c] = sm0 * sm1;
  }
}

// ---------------------------------------------------------------------------
extern "C" void kernel_launch(void* const* d_in, const int* in_sizes, int n_in,
                              void* d_out, int out_size, void* d_ws, size_t ws_size,
                              hipStream_t stream) {
  (void)in_sizes; (void)n_in; (void)out_size; (void)ws_size;
  const float*     x    = (const float*)d_in[0];
  const long long* ei   = (const long long*)d_in[1];   // int64 [2,E]
  const float*     W0   = (const float*)d_in[2];
  // d_in[3] = b0: per-column constant, cancels in softmax(axis=0)
  const float*     g1Wl = (const float*)d_in[4];
  const float*     g1Wr = (const float*)d_in[5];
  const float*     g1al = (const float*)d_in[6];
  const float*     g1ar = (const float*)d_in[7];
  const float*     g1b  = (const float*)d_in[8];
  const float*     g2Wl = (const float*)d_in[9];
  const float*     g2Wr = (const float*)d_in[10];
  const float*     g2al = (const float*)d_in[11];
  const float*     g2ar = (const float*)d_in[12];
  const float*     g2b  = (const float*)d_in[13];
  float* out = (float*)d_out;

  // ---- workspace layout (floats) ----
  float* ws = (float*)d_ws;
  size_t o = 0;
  float*    xl1      = ws + o; o += (size_t)N_NODES * HID;
  float*    h        = ws + o; o += (size_t)N_NODES * HID;
  float*    el1      = ws + o; o += N_NODES;
  float*    er1      = ws + o; o += N_NODES;
  unsigned* emax1Enc = (unsigned*)(ws + o); o += N_NODES;
  float*    emax1    = ws + o; o += N_NODES;
  float*    denom1   = ws + o; o += N_NODES;
  float*    wbuf     = ws + o; o += N_EDGES;
  float*    xl2      = ws + o; o += (size_t)N_NODES * DOUT;
  float*    el2      = ws + o; o += N_NODES;
  float*    er2      = ws + o; o += N_NODES;
  unsigned* emax2Enc = (unsigned*)(ws + o); o += N_NODES;
  float*    emax2    = ws + o; o += N_NODES;
  float*    denom2   = ws + o; o += N_NODES;
  float*    s0       = ws + o; o += (size_t)N_NODES * DOUT;
  float*    s1       = ws + o; o += (size_t)N_NODES * DOUT;
  float*    vr1      = ws + o; o += DIN;
  float*    vr2      = ws + o; o += HID;
  float*    colmax   = ws + o; o += DOUT;
  float*    colsum   = ws + o; o += DOUT;

  auto cdiv = [](long long a, long long b) { return (unsigned)((a + b - 1) / b); };
  const unsigned NEG_INF_ENC = 0x007FFFFFu;  // enc(-inf)

  // 1) fold Wr^T att_r into vectors
  make_vr<<<cdiv(DIN, 256), 256, 0, stream>>>(g1Wr, g1ar, vr1, DIN, HID);
  make_vr<<<cdiv(HID, 256), 256, 0, stream>>>(g2Wr, g2ar, vr2, HID, DOUT);

  // 2) big WMMA GEMMs on x: xl1 = x @ g1Wl^T, s0 = x @ W0^T
  {
    constexpr int CT = HID / 16;             // 8 col tiles
    size_t shm = (size_t)(CT + 1) * DIN * 16 * sizeof(float);   // 288 KB
    wmma_gemm<DIN, CT, HID><<<cdiv(MTILES, 16), CT * 32, shm, stream>>>(
        x, g1Wl, xl1, MTILES, 16);
  }
  {
    constexpr int CT = 2;                    // 20 cols padded to 32
    size_t shm = (size_t)(CT + 1) * DIN * 16 * sizeof(float);   // 96 KB
    wmma_gemm<DIN, CT, DOUT><<<cdiv(MTILES, 5), CT * 32, shm, stream>>>(
        x, W0, s0, MTILES, 5);
  }

  // 3) layer-1 attention scalars
  attn_scalars<<<cdiv(N_NODES, 8), 256, 0, stream>>>(
      x, DIN, vr1, xl1, HID, g1al, el1, er1, N_NODES);

  // 4) layer-1 edge softmax + aggregation into h
  fill_u32<<<cdiv(N_NODES, 256), 256, 0, stream>>>(emax1Enc, NEG_INF_ENC, N_NODES);
  fill_f32<<<cdiv(N_NODES, 256), 256, 0, stream>>>(denom1, 0.0f, N_NODES);
  init_bias<<<cdiv((size_t)N_NODES * HID, 256), 256, 0, stream>>>(
      h, g1b, N_NODES * HID, HID);
  edge_max<<<cdiv(N_EDGES, 256), 256, 0, stream>>>(ei, el1, er1, emax1Enc, N_EDGES);
  decode_max<<<cdiv(N_NODES, 256), 256, 0, stream>>>(emax1Enc, emax1, N_NODES);
  edge_expsum<<<cdiv(N_EDGES, 256), 256, 0, stream>>>(
      ei, el1, er1, emax1, wbuf, denom1, N_EDGES);
  edge_aggregate<<<cdiv((size_t)N_EDGES * HID, 256), 256, 0, stream>>>(
      ei, wbuf, denom1, xl1, h, N_EDGES, HID, HID);
  elu_inplace<<<cdiv((size_t)N_NODES * HID, 256), 256, 0, stream>>>(
      h, N_NODES * HID);

  // 5) layer-2 WMMA GEMM: xl2 = h @ g2Wl^T
  {
    constexpr int CT = 2;
    size_t shm = (size_t)(CT + 1) * HID * 16 * sizeof(float);   // 24 KB
    wmma_gemm<HID, CT, DOUT><<<cdiv(MTILES, 5), CT * 32, shm, stream>>>(
        h, g2Wl, xl2, MTILES, 5);
  }

  // 6) layer-2 attention scalars
  attn_scalars<<<cdiv(N_NODES, 8), 256, 0, stream>>>(
      h, HID, vr2, xl2, DOUT, g2al, el2, er2, N_NODES);

  // 7) layer-2 edge softmax + aggregation into s1 (bias-initialized)
  fill_u32<<<cdiv(N_NODES, 256), 256, 0, stream>>>(emax2Enc, NEG_INF_ENC, N_NODES);
  fill_f32<<<cdiv(N_NODES, 256), 256, 0, stream>>>(denom2, 0.0f, N_NODES);
  init_bias<<<cdiv((size_t)N_NODES * DOUT, 256), 256, 0, stream>>>(
      s1, g2b, N_NODES * DOUT, DOUT);
  edge_max<<<cdiv(N_EDGES, 256), 256, 0, stream>>>(ei, el2, er2, emax2Enc, N_EDGES);
  decode_max<<<cdiv(N_NODES, 256), 256, 0, stream>>>(emax2Enc, emax2, N_NODES);
  edge_expsum<<<cdiv(N_EDGES, 256), 256, 0, stream>>>(
      ei, el2, er2, emax2, wbuf, denom2, N_EDGES);
  edge_aggregate<<<cdiv((size_t)N_EDGES * 32, 256), 256, 0, stream>>>(
      ei, wbuf, denom2, xl2, s1, N_EDGES, DOUT, 32);

  // 8) dual softmax fusion
  col_stats<<<DOUT, 256, 0, stream>>>(s0, colmax, colsum, N_NODES, DOUT);
  final_out<<<cdiv(N_NODES, 256), 256, 0, stream>>>(
      s0, s1, colmax, colsum, out, N_NODES, DOUT);
}